// PACEmodule_64304250356169
// MI455X (gfx1250) — compile-verified
//
#include <hip/hip_runtime.h>

typedef __attribute__((ext_vector_type(2))) float v2f;
typedef __attribute__((ext_vector_type(8))) float v8f;

// ---------------- workspace layout (float offsets) ----------------
constexpr int WS_WSUM = 0;     // 1
constexpr int WS_SW   = 8;     // 64  sqrt(w)
constexpr int WS_BW   = 72;    // 24  b_w[k*3+j]
constexpr int WS_G    = 96;    // 8   g
constexpr int WS_AMAT = 128;   // 1536  A = G @ barB^T   [k*192 + (n*3+i)]
constexpr int WS_H    = 1664;  // 200  h
constexpr int WS_WMAT = 1872;  // 38400 Wmat[(a*3+j)*64 + n] = M[a][n*3+j]
// total 40272 floats (~158 KB) of d_ws

// ================= kernel A: batch-independent precompute =================
__global__ void __launch_bounds__(192) pace_precompute(
    const float* __restrict__ bIn, const float* __restrict__ wIn,
    const float* __restrict__ lamIn, float* __restrict__ ws) {
  __shared__ float sBarB[1536];   // [t*8+k], t = n*3+j
  __shared__ float sA[1536];      // A = G @ barB^T
  __shared__ float sH[64];
  __shared__ float sHinv[64];
  __shared__ float sG[64];
  __shared__ float sg[8];
  __shared__ float sbw[24];
  __shared__ float sScal[1];
  const int tid = threadIdx.x;
  const float lam = lamIn[0];
  const float sqlam = sqrtf(lam);

  if (tid == 0) {
    float s = 0.f;
    for (int n = 0; n < 64; ++n) s += wIn[n];
    ws[WS_WSUM] = s;
    sScal[0] = s;
  }
  __syncthreads();
  const float wsum = sScal[0];
  if (tid < 64) ws[WS_SW + tid] = sqrtf(wIn[tid]);
  if (tid < 24) {
    int k = tid / 3, j = tid - 3 * k;
    float s = 0.f;
    for (int n = 0; n < 64; ++n) s += bIn[k * 192 + j * 64 + n] * wIn[n];
    float v = s / wsum;
    sbw[tid] = v;
    ws[WS_BW + tid] = v;
  }
  __syncthreads();
  for (int i = tid; i < 1536; i += 192) {
    int t = i >> 3, k = i & 7;
    int n = t / 3, j = t - 3 * n;
    sBarB[i] = sqrtf(wIn[n]) * (bIn[k * 192 + j * 64 + n] - sbw[k * 3 + j]);
  }
  __syncthreads();
  if (tid < 64) {
    int k1 = tid >> 3, k2 = tid & 7;
    float s = 0.f;
    for (int t = 0; t < 192; ++t) s += sBarB[t * 8 + k1] * sBarB[t * 8 + k2];
    sH[tid] = 2.f * (s + (k1 == k2 ? lam : 0.f));
  }
  __syncthreads();
  if (tid == 0) {
    // Gauss-Jordan inversion of 8x8 with partial pivoting
    float Am[8][16];
    for (int i = 0; i < 8; ++i)
      for (int j = 0; j < 8; ++j) {
        Am[i][j] = sH[i * 8 + j];
        Am[i][8 + j] = (i == j) ? 1.f : 0.f;
      }
    for (int col = 0; col < 8; ++col) {
      int p = col; float best = fabsf(Am[col][col]);
      for (int r = col + 1; r < 8; ++r) {
        float v = fabsf(Am[r][col]);
        if (v > best) { best = v; p = r; }
      }
      if (p != col)
        for (int c = 0; c < 16; ++c) { float t = Am[col][c]; Am[col][c] = Am[p][c]; Am[p][c] = t; }
      float inv = 1.f / Am[col][col];
      for (int c = 0; c < 16; ++c) Am[col][c] *= inv;
      for (int r = 0; r < 8; ++r) if (r != col) {
        float f = Am[r][col];
        for (int c = 0; c < 16; ++c) Am[r][c] -= f * Am[col][c];
      }
    }
    for (int i = 0; i < 8; ++i)
      for (int j = 0; j < 8; ++j) sHinv[i * 8 + j] = Am[i][8 + j];
    float Ht[8]; float ssum = 0.f;
    for (int k = 0; k < 8; ++k) {
      float s = 0.f;
      for (int j = 0; j < 8; ++j) s += sHinv[k * 8 + j];
      Ht[k] = s; ssum += s;
    }
    for (int k = 0; k < 8; ++k) sg[k] = Ht[k] / ssum;
    for (int k1 = 0; k1 < 8; ++k1)
      for (int k2 = 0; k2 < 8; ++k2)
        sG[k1 * 8 + k2] = sHinv[k1 * 8 + k2] - Ht[k1] * Ht[k2] / ssum;
  }
  __syncthreads();
  if (tid < 8) ws[WS_G + tid] = sg[tid];
  for (int i = tid; i < 1536; i += 192) {
    int k = i / 192, t = i - 192 * k;
    float s = 0.f;
    for (int k2 = 0; k2 < 8; ++k2) s += sG[k * 8 + k2] * sBarB[t * 8 + k2];
    sA[i] = s;
    ws[WS_AMAT + i] = s;
  }
  for (int i = tid; i < 200; i += 192) {
    float hv;
    if (i < 192) {
      float s = 0.f;
      for (int k = 0; k < 8; ++k) s += sBarB[i * 8 + k] * sg[k];
      hv = s;
    } else {
      hv = sqlam * sg[i - 192];
    }
    ws[WS_H + i] = hv;
  }
  __syncthreads();
  // M(200x192) -> Wmat[(a*3+j)*64 + n] = M[a][n*3+j]
  for (int i = tid; i < 600 * 64; i += 192) {
    int a3 = i >> 6, n = i & 63;
    int a = a3 / 3, j = a3 - 3 * a;
    int t2 = n * 3 + j;
    float m;
    if (a < 192) {
      float s = 0.f;
      for (int k = 0; k < 8; ++k) s += sBarB[a * 8 + k] * sA[k * 192 + t2];
      m = 2.f * s - (a == t2 ? 1.f : 0.f);
    } else {
      m = 2.f * sqlam * sA[(a - 192) * 192 + t2];
    }
    ws[WS_WMAT + i] = m;
  }
}

// ================= kernel B: main batched solve =================
// LDS map (floats):
constexpr int FB0 = 0;       // 4 waves * 1280 : F staging [bl][ka(16)][q(16)] ; stage3 overlay: Amat/g/bw
constexpr int YB0 = 5120;    // 4 waves * 1024 : bar_y cols [col(16)][n(64)]
constexpr int QB0 = 9216;    // 120 * 55 packed upper-tri Q
constexpr int HB0 = 15816;   // 200 h
constexpr int SW0 = 16016;   // 64 sqrt(w)
constexpr int WB0 = 16080;   // 64 w
constexpr int YW0 = 16144;   // 4*16 per-wave y_w
constexpr int SMEMF = 16208; // 64,832 bytes

__global__ void __launch_bounds__(128) pace_main(
    const float* __restrict__ yIn, const float* __restrict__ wIn,
    const float* __restrict__ ws, float* __restrict__ out, int Btot) {
  __shared__ float smem[SMEMF];
  const int tid = threadIdx.x;
  const int lane = tid & 31;
  const int wv = tid >> 5;
  const int lo = lane & 15;
  const int hiH = lane >> 4;
  const int ks = hiH * 2;
  const int bBase = blockIdx.x * 120;
  const float* __restrict__ wsW = ws + WS_WMAT;

  for (int i = tid; i < 200; i += 128) smem[HB0 + i] = ws[WS_H + i];
  if (tid < 64) { smem[SW0 + tid] = ws[WS_SW + tid]; smem[WB0 + tid] = wIn[tid]; }
  const float inv_wsum = 1.f / ws[WS_WSUM];
  __syncthreads();

  for (int pass = 0; pass < 6; ++pass) {
    // ---- build bar_y columns for this wave's 5 batches ----
    for (int i = lane; i < 960; i += 32) {
      int bl = i / 192, r = i - 192 * bl;
      int m = r >> 6, n = r & 63;
      int b = bBase + (wv * 6 + pass) * 5 + bl;
      int be = (b < Btot) ? b : (Btot - 1);
      smem[YB0 + wv * 1024 + (bl * 3 + m) * 64 + n] = yIn[(size_t)be * 192 + r];
    }
    for (int i = lane; i < 64; i += 32) smem[YB0 + wv * 1024 + 15 * 64 + i] = 0.f;
    __syncthreads();
    if (lane < 15) {
      float s = 0.f;
      for (int n = 0; n < 64; ++n) s += smem[YB0 + wv * 1024 + lane * 64 + n] * smem[WB0 + n];
      smem[YW0 + wv * 16 + lane] = s * inv_wsum;
    }
    __syncthreads();
    for (int i = lane; i < 960; i += 32) {
      int col = i >> 6, n = i & 63;
      int base = YB0 + wv * 1024 + col * 64 + n;
      smem[base] = smem[SW0 + n] * (smem[base] - smem[YW0 + wv * 16 + col]);
    }
    __syncthreads();

    v8f Qc[5];
#pragma unroll
    for (int bl = 0; bl < 5; ++bl) {
      v8f z = {0.f, 0.f, 0.f, 0.f, 0.f, 0.f, 0.f, 0.f};
      Qc[bl] = z;
    }

    for (int st = 0; st < 13; ++st) {
      // init F staging: col0 = h, rest 0 (covers padding rows a>=200, q=10..15)
      for (int i = lane; i < 1280; i += 32) {
        int q = i & 15;
        int ka = (i >> 4) & 15;
        int a = st * 16 + ka;
        smem[FB0 + wv * 1280 + i] = (q == 0 && a < 200) ? smem[HB0 + a] : 0.f;
      }
      // ---- stage 1: E' tiles,  D(16x16) = Wrows(16x64) x barY(64x16), K=4 steps ----
#pragma unroll
      for (int tt = 0; tt < 3; ++tt) {
        const int a3base = st * 48 + tt * 16;
        if (a3base < 600) {
          const int a3 = a3base + lo;
          const bool rowOK = (a3 < 600);
          const float* wrow = wsW + (size_t)(rowOK ? a3 : 0) * 64 + ks;
          v8f C = {0.f, 0.f, 0.f, 0.f, 0.f, 0.f, 0.f, 0.f};
#pragma unroll
          for (int kk = 0; kk < 16; ++kk) {
            v2f Af;                    // A 16x4: lanes0-15 K=0,1 ; lanes16-31 K=2,3
            Af.x = rowOK ? wrow[kk * 4] : 0.f;
            Af.y = rowOK ? wrow[kk * 4 + 1] : 0.f;
            const int yb = YB0 + wv * 1024 + lo * 64 + kk * 4 + ks;
            v2f Bf;                    // B 4x16 mirrored layout
            Bf.x = smem[yb];
            Bf.y = smem[yb + 1];
            C = __builtin_amdgcn_wmma_f32_16x16x4_f32(false, Af, false, Bf, (short)0, C,
                                                      false, false);
          }
          // scatter C (M=a3 row, N=(batch,m) col) into per-batch F layout
#pragma unroll
          for (int r = 0; r < 8; ++r) {
            const int M = r + 8 * hiH;
            const int a3o = a3base + M;
            if (a3o < 600 && lo < 15) {
              const int a = a3o / 3, j = a3o - 3 * a;
              const int bl = lo / 3, m = lo - 3 * bl;
              const int al = a - st * 16;
              smem[FB0 + wv * 1280 + bl * 256 + al * 16 + 1 + j * 3 + m] = C[r];
            }
          }
        }
      }
      __syncthreads();
      // ---- stage 2: Q += F_chunk^T F_chunk  (A-frag == B-frag) ----
#pragma unroll
      for (int kk2 = 0; kk2 < 4; ++kk2) {
#pragma unroll
        for (int bl = 0; bl < 5; ++bl) {
          const int base = FB0 + wv * 1280 + bl * 256 + (kk2 * 4 + ks) * 16 + lo;
          v2f Ff;
          Ff.x = smem[base];
          Ff.y = smem[base + 16];
          Qc[bl] = __builtin_amdgcn_wmma_f32_16x16x4_f32(false, Ff, false, Ff, (short)0,
                                                         Qc[bl], false, false);
        }
      }
      __syncthreads();
    }

    // park Q (packed upper triangle, 55 floats per batch)
#pragma unroll
    for (int bl = 0; bl < 5; ++bl) {
      const int l = (wv * 6 + pass) * 5 + bl;
#pragma unroll
      for (int r = 0; r < 8; ++r) {
        const int M = r + 8 * hiH;
        const int N = lo;
        if (M <= 9 && N <= 9 && N >= M) {
          const int idx = M * 10 - (M * (M - 1)) / 2 + (N - M);
          smem[QB0 + l * 55 + idx] = Qc[bl][r];
        }
      }
    }
    __syncthreads();
  }

  // ---- stage 3: per-lane batch finish (F/Y regions dead -> overlay A,g,bw) ----
  for (int i = tid; i < 1536; i += 128) smem[FB0 + i] = ws[WS_AMAT + i];
  if (tid < 8) smem[1536 + tid] = ws[WS_G + tid];
  if (tid < 24) smem[1544 + tid] = ws[WS_BW + tid];
  __syncthreads();

  const int l = tid;
  const int b = bBase + l;
  if (l < 120 && b < Btot) {
    float Qm[10][10];
    for (int M = 0; M < 10; ++M)
      for (int N = M; N < 10; ++N) {
        float v = smem[QB0 + l * 55 + M * 10 - (M * (M - 1)) / 2 + (N - M)];
        Qm[M][N] = v;
        Qm[N][M] = v;
      }
    float tr = 0.f;
    for (int d = 0; d < 10; ++d) tr += Qm[d][d];
    const float eps = 1e-6f * tr + 1e-20f;
    for (int d = 0; d < 10; ++d) Qm[d][d] += eps;
    // Cholesky (lower, in place) of PSD Q + eps*I
    for (int i = 0; i < 10; ++i)
      for (int j = 0; j <= i; ++j) {
        float s = Qm[i][j];
        for (int k = 0; k < j; ++k) s -= Qm[i][k] * Qm[j][k];
        if (i == j) Qm[i][j] = sqrtf(fmaxf(s, 1e-30f));
        else Qm[i][j] = s / Qm[j][j];
      }
    // inverse iteration -> eigenvector of smallest eigenvalue
    float x[10], z[10];
    for (int i = 0; i < 10; ++i) x[i] = 1.f;
    for (int it = 0; it < 4; ++it) {
      for (int i = 0; i < 10; ++i) {
        float s = x[i];
        for (int k = 0; k < i; ++k) s -= Qm[i][k] * z[k];
        z[i] = s / Qm[i][i];
      }
      for (int i = 9; i >= 0; --i) {
        float s = z[i];
        for (int k = i + 1; k < 10; ++k) s -= Qm[k][i] * x[k];
        x[i] = s / Qm[i][i];
      }
      float nn = 0.f;
      for (int i = 0; i < 10; ++i) nn += x[i] * x[i];
      const float inv = 1.f / sqrtf(nn + 1e-30f);
      for (int i = 0; i < 10; ++i) x[i] *= inv;
    }
    const float inv0 = 1.f / x[0];
    float R_[3][3];   // R[i][j] = vec[1 + j*3 + i]
    for (int i = 0; i < 3; ++i)
      for (int j = 0; j < 3; ++j) R_[i][j] = x[1 + j * 3 + i] * inv0;
    float ywv[3];
    for (int j = 0; j < 3; ++j) {
      float s = 0.f;
      for (int n = 0; n < 64; ++n) s += yIn[(size_t)b * 192 + j * 64 + n] * smem[WB0 + n];
      ywv[j] = s * inv_wsum;
    }
    float cvec[8];
    for (int k = 0; k < 8; ++k) cvec[k] = smem[1536 + k];   // g
    for (int n = 0; n < 64; ++n) {
      float by[3];
      for (int j = 0; j < 3; ++j)
        by[j] = smem[SW0 + n] * (yIn[(size_t)b * 192 + j * 64 + n] - ywv[j]);
      for (int i = 0; i < 3; ++i) {
        const float Bvi = R_[0][i] * by[0] + R_[1][i] * by[1] + R_[2][i] * by[2];
        const int t2 = n * 3 + i;
        for (int k = 0; k < 8; ++k) cvec[k] += 2.f * smem[FB0 + k * 192 + t2] * Bvi;
      }
    }
    float u[3];
    for (int i = 0; i < 3; ++i) {
      float s = 0.f;
      for (int k = 0; k < 8; ++k) s += smem[1544 + k * 3 + i] * cvec[k];
      u[i] = s;
    }
    for (int i = 0; i < 3; ++i)
      for (int j = 0; j < 3; ++j) out[(size_t)b * 9 + i * 3 + j] = R_[i][j];
    for (int j = 0; j < 3; ++j)
      out[(size_t)Btot * 9 + (size_t)b * 3 + j] =
          ywv[j] - (R_[j][0] * u[0] + R_[j][1] * u[1] + R_[j][2] * u[2]);
    for (int k = 0; k < 8; ++k)
      out[(size_t)Btot * 12 + (size_t)b * 8 + k] = cvec[k];
  }
}

extern "C" void kernel_launch(void* const* d_in, const int* in_sizes, int n_in,
                              void* d_out, int out_size, void* d_ws, size_t ws_size,
                              hipStream_t stream) {
  const float* y = (const float*)d_in[0];
  const float* bb = (const float*)d_in[1];
  const float* w = (const float*)d_in[2];
  const float* lam = (const float*)d_in[3];
  float* ws = (float*)d_ws;
  float* out = (float*)d_out;
  const int Btot = in_sizes[0] / 192;  // y is (B,3,64)
  pace_precompute<<<1, 192, 0, stream>>>(bb, w, lam, ws);
  const int blocks = (Btot + 119) / 120;
  pace_main<<<blocks, 128, 0, stream>>>(y, w, ws, out, Btot);
  (void)n_in; (void)out_size; (void)ws_size;
}